// Grouped_Soft_GSL_Block_PE_54829552501126
// MI455X (gfx1250) — compile-verified
//
#include <hip/hip_runtime.h>
#include <math.h>

// ---------------------------------------------------------------------------
// Fused Grouped_Soft_GSL block for MI455X (gfx1250), wave32, WMMA f32 16x16x4.
// Memory-bound (x = 268MB, AI ~64 F/B) -> read x once, all GEMMs via WMMA,
// weights pre-folded+pre-swizzled into WMMA A-register order (L2-resident).
// ---------------------------------------------------------------------------

typedef float v2f __attribute__((ext_vector_type(2)));
typedef float v8f __attribute__((ext_vector_type(8)));

#define DEVI __device__ __forceinline__

// problem constants
constexpr int BB = 8, IC2 = 128, NN = 2048, KK = 32, CC = 64;
constexpr int NK = NN * KK;                       // 65536 pixels per (b,channel)
constexpr long long XBATCH = (long long)IC2 * NK; // x per-batch stride (floats)

// ---- workspace layout (floats) --------------------------------------------
// All "A" arrays are stored in WMMA A-operand register order:
//   entry[(tile*KSTEPS + ks)*32 + lane] is a float2 = A[row][col], A[row][col+1]
//   with row = tile*16 + (lane&15), col = ks*4 + 2*(lane>>4).
constexpr int OFF_WMAIN = 0;                       // [6][32][32][2] trans|gate1|attn1
constexpr int OFF_WPOS1 = OFF_WMAIN + 6 * 32 * 64; // 12288: [4][1][32][2] pos l1 (K=4, col3=0)
constexpr int OFF_WPOS2 = OFF_WPOS1 + 4 * 64;      // 12544: [4][16][32][2] pos l2
constexpr int OFF_WPE   = OFF_WPOS2 + 4 * 16 * 64; // 16640: [1][16][32][2] pe_proj (rows 8..15 = 0)
constexpr int OFF_WATT2 = OFF_WPE + 16 * 64;       // 17664: [1][4][32][2]  attn_w2 (rows 8..15 = 0)
constexpr int OFF_BIAS  = OFF_WATT2 + 4 * 64;      // 17920
// bias block (relative offsets)
constexpr int BO_POS1 = 0, BO_POS2 = 64, BO_TRANS = 128, BO_G1 = 192,
              BO_A1 = 208, BO_GW2 = 224, BO_GB2 = 240, BO_AB2 = 241, BO_PEB = 249;
constexpr int BIAS_N = 257;
constexpr int WS_FLOATS = OFF_BIAS + BIAS_N;       // ~72.7 KB

// ---------------------------------------------------------------------------
// Prep kernel: fold BN into weights/biases, swizzle into WMMA A order.
// ---------------------------------------------------------------------------
__global__ void gsl_prep(
    const float* __restrict__ pos_w1, const float* __restrict__ pg,
    const float* __restrict__ pb, const float* __restrict__ pm, const float* __restrict__ pv,
    const float* __restrict__ pos_w2, const float* __restrict__ pos_b2,
    const float* __restrict__ pe_w, const float* __restrict__ pe_b,
    const float* __restrict__ trans_w, const float* __restrict__ tg,
    const float* __restrict__ tb, const float* __restrict__ tm, const float* __restrict__ tv,
    const float* __restrict__ gate_w1, const float* __restrict__ gg,
    const float* __restrict__ gb, const float* __restrict__ gm, const float* __restrict__ gv,
    const float* __restrict__ gate_w2, const float* __restrict__ gate_b2,
    const float* __restrict__ attn_w1, const float* __restrict__ ag,
    const float* __restrict__ ab, const float* __restrict__ am, const float* __restrict__ av,
    const float* __restrict__ attn_w2, const float* __restrict__ attn_b2,
    float* __restrict__ ws)
{
    int idx = blockIdx.x * blockDim.x + threadIdx.x;
    if (idx >= WS_FLOATS) return;
    float val = 0.f;
    if (idx < OFF_WPOS1) {                     // fused main GEMM A: 96 rows x 128 cols
        int t = idx; int j = t & 1; int l = (t >> 1) & 31; int ks = (t >> 6) & 31; int ct = t >> 11;
        int row = ct * 16 + (l & 15);
        int col = ks * 4 + 2 * (l >> 4) + j;
        if (row < 64)      val = trans_w[row * IC2 + col] * (tg[row] * rsqrtf(tv[row] + 1e-5f));
        else if (row < 80) { int r = row - 64; val = gate_w1[r * IC2 + col] * (gg[r] * rsqrtf(gv[r] + 1e-5f)); }
        else               { int r = row - 80; val = attn_w1[r * IC2 + col] * (ag[r] * rsqrtf(av[r] + 1e-5f)); }
    } else if (idx < OFF_WPOS2) {              // pos layer1: 64x3 padded to K=4
        int t = idx - OFF_WPOS1; int j = t & 1; int l = (t >> 1) & 31; int ct = t >> 6;
        int row = ct * 16 + (l & 15); int col = 2 * (l >> 4) + j;
        val = (col < 3) ? pos_w1[row * 3 + col] * (pg[row] * rsqrtf(pv[row] + 1e-5f)) : 0.f;
    } else if (idx < OFF_WPE) {                // pos layer2: 64x64 (bias-only, no BN)
        int t = idx - OFF_WPOS2; int j = t & 1; int l = (t >> 1) & 31; int ks = (t >> 6) & 15; int ct = t >> 10;
        int row = ct * 16 + (l & 15); int col = ks * 4 + 2 * (l >> 4) + j;
        val = pos_w2[row * CC + col];
    } else if (idx < OFF_WATT2) {              // pe_proj: 8x64 padded to 16 rows
        int t = idx - OFF_WPE; int j = t & 1; int l = (t >> 1) & 31; int ks = t >> 6;
        int row = l & 15; int col = ks * 4 + 2 * (l >> 4) + j;
        val = (row < 8) ? pe_w[row * CC + col] : 0.f;
    } else if (idx < OFF_BIAS) {               // attn_w2: 8x16 padded to 16 rows
        int t = idx - OFF_WATT2; int j = t & 1; int l = (t >> 1) & 31; int ks = t >> 6;
        int row = l & 15; int col = ks * 4 + 2 * (l >> 4) + j;
        val = (row < 8) ? attn_w2[row * 16 + col] : 0.f;
    } else {                                   // folded biases + small vectors
        int t = idx - OFF_BIAS;
        if (t < 64)       val = pb[t] - pm[t] * (pg[t] * rsqrtf(pv[t] + 1e-5f));
        else if (t < 128) val = pos_b2[t - 64];
        else if (t < 192) { int r = t - 128; val = tb[r] - tm[r] * (tg[r] * rsqrtf(tv[r] + 1e-5f)); }
        else if (t < 208) { int r = t - 192; val = gb[r] - gm[r] * (gg[r] * rsqrtf(gv[r] + 1e-5f)); }
        else if (t < 224) { int r = t - 208; val = ab[r] - am[r] * (ag[r] * rsqrtf(av[r] + 1e-5f)); }
        else if (t < 240) val = gate_w2[t - 224];
        else if (t == 240) val = gate_b2[0];
        else if (t < 249) val = attn_b2[t - 241];
        else              val = pe_b[t - 249];
    }
    ws[idx] = val;
}

// ---------------------------------------------------------------------------
// Main fused kernel. 128 threads = 4 waves; each wave owns one point n
// (32 neighbors -> 2 WMMA pixel-tiles of 16). Pixels ride the WMMA N dim.
// ---------------------------------------------------------------------------
DEVI float lrelu(float x) { return x > 0.f ? x : 0.2f * x; }

DEVI v8f wmma4(v2f a, v2f b, v8f c) {
    // 8-arg form: (neg_a, A, neg_b, B, c_mod, C, reuse_a, reuse_b)
    return __builtin_amdgcn_wmma_f32_16x16x4_f32(false, a, false, b, (short)0, c, false, false);
}

// LDS tile row stride: 20 floats -> half-wave bank sets (offset 8*20=160 = 32 mod 64)
// are disjoint for both the D-layout writes and B-layout reads.
constexpr int LSTR = 20;

__global__ __launch_bounds__(128) void gsl_main(
    const float* __restrict__ x, const float* __restrict__ rel,
    const float* __restrict__ ws, float* __restrict__ out)
{
    __shared__ float sHid[4][64 * LSTR];   // per-wave: pos_hidden, then attn_h scratch
    __shared__ float sEnc[4][64 * LSTR];   // per-wave: pos_enc (transposed source for pe/trans-add)
    __shared__ float sB[BIAS_N];

    const int tid = threadIdx.x;
    for (int i = tid; i < BIAS_N; i += 128) sB[i] = ws[OFF_BIAS + i];
    __syncthreads();

    const int wid  = tid >> 5;
    const int lane = tid & 31;
    const int half = lane >> 4;      // which K-pair of the A/B operands this lane holds
    const int p    = lane & 15;      // pixel within 16-wide tile (WMMA N dim)

    const int pid = blockIdx.x * 4 + wid;      // flat point id in [0, B*N)
    const int b   = pid >> 11;                 // 2048 points per batch
    const int n   = pid & 2047;

    float* HID = sHid[wid];
    float* ENC = sEnc[wid];
    const v2f* Wmain = (const v2f*)(ws + OFF_WMAIN);
    const v2f* Wpos1 = (const v2f*)(ws + OFF_WPOS1);
    const v2f* Wpos2 = (const v2f*)(ws + OFF_WPOS2);
    const v2f* Wpe   = (const v2f*)(ws + OFF_WPE);
    const v2f* Watt2 = (const v2f*)(ws + OFF_WATT2);

    float tf[2][4][8];     // trans_feat tiles (D layout), kept in registers
    float gate[2];         // sigmoid gate per pixel (valid in both lane halves)
    float lgt[2][8];       // attn logits per group (valid in lanes 0..15)

#pragma unroll
    for (int pt = 0; pt < 2; ++pt) {
        const long long pixbase = (long long)n * KK + pt * 16 + p;

        // ---- pos_net layer1 (3->64, BN folded, lrelu), K=4 with zero pad ----
        // B-operand layout assumption (mirrors documented A layout): lane half
        // selects K rows {0,1} vs {2,3}; vgpr0/1 = those two rows at pixel p.
        {
            const long long rbase = (long long)b * 3 * NK + pixbase;
            int c0 = 2 * half;
            float rb0 = rel[rbase + (long long)c0 * NK];                // c0 in {0,2} < 3
            float rb1 = (c0 + 1 < 3) ? rel[rbase + (long long)(c0 + 1) * NK] : 0.f;
            v2f rB = {rb0, rb1};
#pragma unroll
            for (int ct = 0; ct < 4; ++ct) {
                v8f h = {};
                h = wmma4(Wpos1[ct * 32 + lane], rB, h);
#pragma unroll
                for (int r = 0; r < 8; ++r) {
                    int ch = ct * 16 + r + 8 * half;   // C/D layout: lanes16-31 hold M+8
                    HID[ch * LSTR + p] = lrelu(h[r] + sB[BO_POS1 + ch]);
                }
            }
        }

        // ---- pos_net layer2 (64->64 + bias) -> pos_enc (LDS, transposed) ----
#pragma unroll
        for (int ct = 0; ct < 4; ++ct) {
            v8f e = {};
#pragma unroll 4
            for (int ks = 0; ks < 16; ++ks) {
                int k0 = ks * 4 + 2 * half;
                v2f bb = {HID[k0 * LSTR + p], HID[(k0 + 1) * LSTR + p]};
                e = wmma4(Wpos2[(ct * 16 + ks) * 32 + lane], bb, e);
            }
#pragma unroll
            for (int r = 0; r < 8; ++r) {
                int ch = ct * 16 + r + 8 * half;
                ENC[ch * LSTR + p] = e[r] + sB[BO_POS2 + ch];
            }
        }

        // ---- pe_proj (8<-64, padded to 16 rows) on pos_enc ----
        v8f pe = {};
#pragma unroll 4
        for (int ks = 0; ks < 16; ++ks) {
            int k0 = ks * 4 + 2 * half;
            v2f bb = {ENC[k0 * LSTR + p], ENC[(k0 + 1) * LSTR + p]};
            pe = wmma4(Wpe[ks * 32 + lane], bb, pe);
        }

        // ---- fused main GEMM: [trans64 | gate_h16 | attn_h16] x K=128 ----
        // x read exactly once per pixel per channel (HBM-bound path).
        v8f acc[6] = {v8f{}, v8f{}, v8f{}, v8f{}, v8f{}, v8f{}};
        const float* xb = x + (long long)b * XBATCH + pixbase;
#pragma unroll 4
        for (int ks = 0; ks < 32; ++ks) {
            int ch0 = ks * 4 + 2 * half;
            v2f xB = {xb[(long long)ch0 * NK], xb[(long long)(ch0 + 1) * NK]};
#pragma unroll
            for (int ct = 0; ct < 6; ++ct)
                acc[ct] = wmma4(Wmain[(ct * 32 + ks) * 32 + lane], xB, acc[ct]);
        }

        // trans tiles: BN bias + lrelu + pos_enc (same D layout -> LDS gather)
#pragma unroll
        for (int ct = 0; ct < 4; ++ct)
#pragma unroll
            for (int r = 0; r < 8; ++r) {
                int ch = ct * 16 + r + 8 * half;
                tf[pt][ct][r] = lrelu(acc[ct][r] + sB[BO_TRANS + ch]) + ENC[ch * LSTR + p];
            }

        // gate: dot(gate_w2, lrelu(gate_h)) + b2 -> sigmoid (hidden dim is
        // split across lane halves -> one xor-16 combine gives both halves)
        float gl = 0.f;
#pragma unroll
        for (int r = 0; r < 8; ++r) {
            int hch = r + 8 * half;
            gl += sB[BO_GW2 + hch] * lrelu(acc[4][r] + sB[BO_G1 + hch]);
        }
        gl += __shfl_xor(gl, 16, 32);
        gl += sB[BO_GB2];
        float g = 1.f / (1.f + __expf(-gl));
        gate[pt] = g;
        float lg = __logf(g + 1e-6f);

        // attn logits: transpose attn_h through LDS (reuse HID), 16x16 WMMA
#pragma unroll
        for (int r = 0; r < 8; ++r) {
            int hch = r + 8 * half;
            HID[hch * LSTR + p] = lrelu(acc[5][r] + sB[BO_A1 + hch]);
        }
        v8f al = {};
#pragma unroll
        for (int ks = 0; ks < 4; ++ks) {
            int k0 = ks * 4 + 2 * half;
            v2f bb = {HID[k0 * LSTR + p], HID[(k0 + 1) * LSTR + p]};
            al = wmma4(Watt2[ks * 32 + lane], bb, al);
        }
        // groups live in lanes 0..15 (rows 0..7); lanes 16..31 carry padding.
#pragma unroll
        for (int r = 0; r < 8; ++r)
            lgt[pt][r] = al[r] + pe[r] + sB[BO_AB2 + r] + sB[BO_PEB + r] + lg;
    }

    // ---- softmax over the 32 neighbors (2 tiles x 16 lanes) per group ----
    float w0[8], w1[8];
#pragma unroll
    for (int r = 0; r < 8; ++r) {
        float m = fmaxf(lgt[0][r], lgt[1][r]);
        m = fmaxf(m, __shfl_xor(m, 1, 32));
        m = fmaxf(m, __shfl_xor(m, 2, 32));
        m = fmaxf(m, __shfl_xor(m, 4, 32));
        m = fmaxf(m, __shfl_xor(m, 8, 32));
        float e0 = __expf(lgt[0][r] - m);
        float e1 = __expf(lgt[1][r] - m);
        float s = e0 + e1;
        s += __shfl_xor(s, 1, 32);
        s += __shfl_xor(s, 2, 32);
        s += __shfl_xor(s, 4, 32);
        s += __shfl_xor(s, 8, 32);
        float inv = 1.f / s;
        w0[r] = e0 * inv; w1[r] = e1 * inv;
        // broadcast valid lower-half weights into the upper half
        float u0 = __shfl_xor(w0[r], 16, 32);
        float u1 = __shfl_xor(w1[r], 16, 32);
        if (half) { w0[r] = u0; w1[r] = u1; }
    }

    // ---- out = sum_k trans*attn_w(group)  +  max_k trans*gate ----
#pragma unroll
    for (int ct = 0; ct < 4; ++ct) {
        // channel ct*16 + r + 8*half -> group 2*ct + half (CPG = 8)
        float sg0 = half ? w0[2 * ct + 1] : w0[2 * ct];
        float sg1 = half ? w1[2 * ct + 1] : w1[2 * ct];
#pragma unroll
        for (int r = 0; r < 8; ++r) {
            float asum = tf[0][ct][r] * sg0 + tf[1][ct][r] * sg1;
            asum += __shfl_xor(asum, 1, 32);
            asum += __shfl_xor(asum, 2, 32);
            asum += __shfl_xor(asum, 4, 32);
            asum += __shfl_xor(asum, 8, 32);
            float mv = fmaxf(tf[0][ct][r] * gate[0], tf[1][ct][r] * gate[1]);
            mv = fmaxf(mv, __shfl_xor(mv, 1, 32));
            mv = fmaxf(mv, __shfl_xor(mv, 2, 32));
            mv = fmaxf(mv, __shfl_xor(mv, 4, 32));
            mv = fmaxf(mv, __shfl_xor(mv, 8, 32));
            if (p == 0) {
                int ch = ct * 16 + r + 8 * half;   // lanes 0 and 16 write
                out[((long long)b * CC + ch) * NN + n] = asum + mv;
            }
        }
    }
}

// ---------------------------------------------------------------------------
extern "C" void kernel_launch(void* const* d_in, const int* in_sizes, int n_in,
                              void* d_out, int out_size, void* d_ws, size_t ws_size,
                              hipStream_t stream) {
    const float* x        = (const float*)d_in[0];
    const float* rel_pos  = (const float*)d_in[1];
    const float* pos_w1   = (const float*)d_in[2];
    const float* pos_bn_g = (const float*)d_in[3];
    const float* pos_bn_b = (const float*)d_in[4];
    const float* pos_bn_m = (const float*)d_in[5];
    const float* pos_bn_v = (const float*)d_in[6];
    const float* pos_w2   = (const float*)d_in[7];
    const float* pos_b2   = (const float*)d_in[8];
    const float* pe_w     = (const float*)d_in[9];
    const float* pe_b     = (const float*)d_in[10];
    const float* trans_w  = (const float*)d_in[11];
    const float* trans_g  = (const float*)d_in[12];
    const float* trans_b  = (const float*)d_in[13];
    const float* trans_m  = (const float*)d_in[14];
    const float* trans_v  = (const float*)d_in[15];
    const float* gate_w1  = (const float*)d_in[16];
    const float* gate_g   = (const float*)d_in[17];
    const float* gate_b   = (const float*)d_in[18];
    const float* gate_m   = (const float*)d_in[19];
    const float* gate_v   = (const float*)d_in[20];
    const float* gate_w2  = (const float*)d_in[21];
    const float* gate_b2  = (const float*)d_in[22];
    const float* attn_w1  = (const float*)d_in[23];
    const float* attn_g   = (const float*)d_in[24];
    const float* attn_b   = (const float*)d_in[25];
    const float* attn_m   = (const float*)d_in[26];
    const float* attn_v   = (const float*)d_in[27];
    const float* attn_w2  = (const float*)d_in[28];
    const float* attn_b2  = (const float*)d_in[29];

    float* ws = (float*)d_ws;

    gsl_prep<<<(WS_FLOATS + 255) / 256, 256, 0, stream>>>(
        pos_w1, pos_bn_g, pos_bn_b, pos_bn_m, pos_bn_v, pos_w2, pos_b2,
        pe_w, pe_b,
        trans_w, trans_g, trans_b, trans_m, trans_v,
        gate_w1, gate_g, gate_b, gate_m, gate_v, gate_w2, gate_b2,
        attn_w1, attn_g, attn_b, attn_m, attn_v, attn_w2, attn_b2,
        ws);

    gsl_main<<<(BB * NN) / 4, 128, 0, stream>>>(x, rel_pos, ws, (float*)d_out);
}